// QuanvolutionGen_65481071399360
// MI455X (gfx1250) — compile-verified
//
#include <hip/hip_runtime.h>
#include <math.h>

typedef float v2f __attribute__((ext_vector_type(2)));
typedef float v8f __attribute__((ext_vector_type(8)));

#define DIMQ 16
#define WAVES_PER_BLOCK 8

// ---------------------------------------------------------------------------
// Kernel 1: build the fixed 16x16 entangling unitary U = prod_i (CN_i RY_i CN_i)
// from ent_w (device input), store row-major to d_ws. One thread; trivial cost.
// ---------------------------------------------------------------------------
__global__ void build_U_kernel(const float* __restrict__ ent_w,
                               float* __restrict__ Uout) {
    if (threadIdx.x != 0 || blockIdx.x != 0) return;
    float U[DIMQ][DIMQ];
    for (int r = 0; r < DIMQ; ++r)
        for (int c = 0; c < DIMQ; ++c)
            U[r][c] = (r == c) ? 1.0f : 0.0f;

    for (int i = 0; i < 4; ++i) {
        const int control = i;
        const int target  = (i + 1) & 3;
        const int cmask = 1 << (3 - control);   // wire 0 = MSB
        const int tmask = 1 << (3 - target);
        const float ch = __cosf(0.5f * ent_w[i]);
        const float sh = __sinf(0.5f * ent_w[i]);
        const int rymask = 1 << (3 - i);

        // CNOT: U'[r,:] = U[perm(r),:]  (swap row pairs where control bit set)
        for (int r = 0; r < DIMQ; ++r) {
            if ((r & cmask) && !(r & tmask)) {
                const int r2 = r | tmask;
                for (int c = 0; c < DIMQ; ++c) {
                    float t = U[r][c]; U[r][c] = U[r2][c]; U[r2][c] = t;
                }
            }
        }
        // RY on wire i: rows (b=0, b=1) mix with [[c,-s],[s,c]]
        for (int r = 0; r < DIMQ; ++r) {
            if (!(r & rymask)) {
                const int r1 = r | rymask;
                for (int c = 0; c < DIMQ; ++c) {
                    float u0 = U[r][c], u1 = U[r1][c];
                    U[r][c]  = ch * u0 - sh * u1;
                    U[r1][c] = sh * u0 + ch * u1;
                }
            }
        }
        // CNOT again
        for (int r = 0; r < DIMQ; ++r) {
            if ((r & cmask) && !(r & tmask)) {
                const int r2 = r | tmask;
                for (int c = 0; c < DIMQ; ++c) {
                    float t = U[r][c]; U[r][c] = U[r2][c]; U[r2][c] = t;
                }
            }
        }
    }
    for (int r = 0; r < DIMQ; ++r)
        for (int c = 0; c < DIMQ; ++c)
            Uout[r * DIMQ + c] = U[r][c];
}

// ---------------------------------------------------------------------------
// Kernel 2: one wave32 per batch element. Phi = U @ Psi via v_wmma_f32_16x16x4_f32,
// amp = sum(phi^2, m<8) - sum(phi^2, m>=8); head + log_softmax per wave.
// ---------------------------------------------------------------------------
__global__ __launch_bounds__(256) void quanv_kernel(
        const float* __restrict__ x,        // [B,784]
        const float* __restrict__ head_w,   // [10,392]
        const float* __restrict__ head_b,   // [10]
        const float* __restrict__ Umat,     // [16,16] row-major (d_ws)
        float* __restrict__ out) {          // [B,10]

    __shared__ __align__(16) float imgLds[WAVES_PER_BLOCK][784];
    __shared__ float featLds[WAVES_PER_BLOCK][196];
    __shared__ float W2lds[10 * 196];       // head_w[c, 2p] compacted

    const int tid  = threadIdx.x;
    const int wave = tid >> 5;
    const int lane = tid & 31;
    const int hi   = lane >> 4;             // 0: lanes 0-15, 1: lanes 16-31
    const int col  = lane & 15;             // B/D column (patch) and A row (state)

    // Cooperative staging of compacted head weights (odd features are 0).
    for (int i = tid; i < 10 * 196; i += 256) {
        const int c = i / 196;
        const int p = i - c * 196;
        W2lds[i] = head_w[c * 392 + 2 * p];
    }

    const int b = blockIdx.x * WAVES_PER_BLOCK + wave;

    // Stage this wave's image (784 floats = 196 float4, 16B-aligned rows).
    const float4* img4 = (const float4*)(x + (size_t)b * 784);
    float4* dst4 = (float4*)(&imgLds[wave][0]);
    for (int t = lane; t < 196; t += 32) dst4[t] = img4[t];

    // A-operand (U) registers: lane holds row M=col; VGPR0=K(4k + hi?2:0),
    // VGPR1=K(4k + hi?3:1) per the 32-bit 16x4 A layout.
    v2f ua[4];
#pragma unroll
    for (int kk = 0; kk < 4; ++kk) {
        ua[kk][0] = Umat[col * 16 + 4 * kk + (hi ? 2 : 0)];
        ua[kk][1] = Umat[col * 16 + 4 * kk + (hi ? 3 : 1)];
    }

    __syncthreads();

    const float* img = &imgLds[wave][0];

    // 13 tiles of 16 patches (196 patches, tail masked to psi=0).
    for (int tt = 0; tt < 13; ++tt) {
        const int p     = tt * 16 + col;
        const int valid = (p < 196);
        const int p2    = valid ? p : 0;
        const int pr = p2 / 14;
        const int pc = p2 - pr * 14;
        const int r0 = 2 * pr, c0 = 2 * pc;

        const float a0 = img[r0 * 28 + c0];
        const float a1 = img[r0 * 28 + c0 + 1];
        const float a2 = img[(r0 + 1) * 28 + c0];
        const float a3 = img[(r0 + 1) * 28 + c0 + 1];

        const float c0q = __cosf(0.5f * a0), s0q = __sinf(0.5f * a0);
        const float c1q = __cosf(0.5f * a1), s1q = __sinf(0.5f * a1);
        const float c2q = __cosf(0.5f * a2), s2q = __sinf(0.5f * a2);
        const float c3q = __cosf(0.5f * a3), s3q = __sinf(0.5f * a3);

        // psi[m] = q0[b0] q1[b1] q2[b2] q3[b3], m = (b0<<3)|(b1<<2)|(b2<<1)|b3.
        // Row block kk covers m = 4kk..4kk+3: A_kk * (q2 q3 combos).
        const float A00 = c0q * c1q, A01 = c0q * s1q;
        const float A10 = s0q * c1q, A11 = s0q * s1q;
        const float msk = valid ? 1.0f : 0.0f;
        const float Bx = (hi ? s2q * c3q : c2q * c3q) * msk;  // b3=0 element
        const float By = (hi ? s2q * s3q : c2q * s3q) * msk;  // b3=1 element

        v2f b0v; b0v[0] = A00 * Bx; b0v[1] = A00 * By;
        v2f b1v; b1v[0] = A01 * Bx; b1v[1] = A01 * By;
        v2f b2v; b2v[0] = A10 * Bx; b2v[1] = A10 * By;
        v2f b3v; b3v[0] = A11 * Bx; b3v[1] = A11 * By;

        v8f acc = {};
        acc = __builtin_amdgcn_wmma_f32_16x16x4_f32(false, ua[0], false, b0v,
                                                    (short)0, acc, false, false);
        acc = __builtin_amdgcn_wmma_f32_16x16x4_f32(false, ua[1], false, b1v,
                                                    (short)0, acc, false, false);
        acc = __builtin_amdgcn_wmma_f32_16x16x4_f32(false, ua[2], false, b2v,
                                                    (short)0, acc, false, false);
        acc = __builtin_amdgcn_wmma_f32_16x16x4_f32(false, ua[3], false, b3v,
                                                    (short)0, acc, false, false);

        // D layout: lane n holds phi rows 0..7, lane n+16 rows 8..15 of column n.
        float ss = 0.0f;
#pragma unroll
        for (int r = 0; r < 8; ++r) ss = fmaf(acc[r], acc[r], ss);
        const float sgn = hi ? -ss : ss;
        const float amp = sgn + __shfl_xor(sgn, 16, 32);

        if (!hi && valid) featLds[wave][p] = amp;
    }

    __syncthreads();

    // Head: logits[c] = head_b[c] + sum_p amp_p * W2[c,p]; then log_softmax.
    float lg = -1e30f;
    if (lane < 10) {
        float accv = head_b[lane];
        const float* w = &W2lds[lane * 196];
        const float* f = &featLds[wave][0];
        for (int p = 0; p < 196; ++p) accv = fmaf(f[p], w[p], accv);
        lg = accv;
    }
    float m = lg;
#pragma unroll
    for (int off = 1; off <= 8; off <<= 1)
        m = fmaxf(m, __shfl_xor(m, off, 32));
    float e = (lane < 10) ? expf(lg - m) : 0.0f;
    float ssum = e;
#pragma unroll
    for (int off = 1; off <= 8; off <<= 1)
        ssum += __shfl_xor(ssum, off, 32);
    const float outv = lg - m - logf(ssum);
    if (lane < 10) out[(size_t)b * 10 + lane] = outv;
}

// ---------------------------------------------------------------------------
extern "C" void kernel_launch(void* const* d_in, const int* in_sizes, int n_in,
                              void* d_out, int out_size, void* d_ws, size_t ws_size,
                              hipStream_t stream) {
    const float* x      = (const float*)d_in[0];   // [B,1,28,28]
    const float* ent_w  = (const float*)d_in[1];   // [4]
    const float* head_w = (const float*)d_in[2];   // [10,392]
    const float* head_b = (const float*)d_in[3];   // [10]
    float* out = (float*)d_out;
    float* U   = (float*)d_ws;                     // 256 floats

    const int B = in_sizes[0] / 784;               // 4096

    build_U_kernel<<<1, 32, 0, stream>>>(ent_w, U);
    quanv_kernel<<<B / WAVES_PER_BLOCK, 256, 0, stream>>>(x, head_w, head_b, U, out);
}